// ERA5_pressure_11407433138733
// MI455X (gfx1250) — compile-verified
//
#include <hip/hip_runtime.h>

#define NXg 1440
#define NYg 721
#define NNg (NXg * NYg)
#define LXg (2.0f * 3.14159265358979323846f * 6371000.0f)
#define NU  0.001f

typedef __attribute__((ext_vector_type(2))) float v2f;
typedef __attribute__((ext_vector_type(8))) float v8f;

static __device__ __forceinline__ int iclampi(int v, int lo, int hi) {
    return v < lo ? lo : (v > hi ? hi : v);
}

// v_rcp_f32 + one Newton-Raphson step (~1 ulp), ~3 VALU instead of ~10 for IEEE div.
static __device__ __forceinline__ float fastrcp(float x) {
    float r = __builtin_amdgcn_rcpf(x);
    r = r * __builtin_fmaf(-x, r, 2.0f);
    return r;
}

#define WMMA4(A, B, C) \
    __builtin_amdgcn_wmma_f32_16x16x4_f32(false, (A), false, (B), (short)0, (C), false, false)

__global__ __launch_bounds__(256) void era5_tend_kernel(
    const float* __restrict__ U, const float* __restrict__ V, const float* __restrict__ P,
    const float* __restrict__ X, const float* __restrict__ Y,
    const float* __restrict__ MF, const float* __restrict__ FC,
    float* __restrict__ OUT)
{
    const int TJ = (NYg + 15) / 16;   // 46 tiles along y
    const int TI = NXg / 16;          // 90 tiles along x
    const int NT = TI * TJ;           // 4140

    // Per-wave coefficient tables in LDS (8 waves / block).
    __shared__ int   oS [8][16];      // band base offset per output j (0 / -1 / -2)
    __shared__ float ycS[8][16][6];   // a1,b1,c1,a2,b2,c2 (ascending jk within band)
    __shared__ float xcS[8][16][6];   // a1x,b1x,c1x,a2x,b2x,c2x per row i

    int wid = blockIdx.x * 8 + (threadIdx.x >> 5);  // 8 waves / block
    if (wid >= NT) return;                          // wave-uniform exit

    int lane = threadIdx.x & 31;
    int wv   = threadIdx.x >> 5;
    int hi   = lane >> 4;       // half-wave select
    int mn   = lane & 15;       // A: row m; B/D: col n
    int tj = wid % TJ, ti = wid / TJ;
    int j0 = tj * 16, i0 = ti * 16;

    // ---------- build per-wave coefficient tables (3 reciprocals per lane) ----------
    if (lane < 16) {
        // y-direction coefficients for output jn = j0 + lane (boundary formulas folded)
        int jn_ = j0 + lane;
        int o = 0;
        float a1 = 0.f, b1 = 0.f, c1 = 0.f, a2 = 0.f, b2 = 0.f, c2 = 0.f;
        if (jn_ < NYg) {
            if (jn_ == 0) {
                float h1 = Y[1] - Y[0], h2 = Y[2] - Y[1];
                float p1 = fastrcp(h1 * (h1 + h2));
                float p2 = fastrcp(h1 * h2);
                float p3 = fastrcp(h2 * (h1 + h2));
                o  = 0;                       // band covers jk = 0,1,2
                a1 = -(2.0f * h1 + h2) * p1;  b1 = (h1 + h2) * p2;  c1 = -h1 * p3;
                a2 = 2.0f * p1;               b2 = -2.0f * p2;      c2 = 2.0f * p3;
            } else if (jn_ == NYg - 1) {
                float g1 = Y[NYg - 1] - Y[NYg - 2], g2 = Y[NYg - 2] - Y[NYg - 3];
                float r1 = fastrcp(g2 * (g1 + g2));
                float r2 = fastrcp(g2 * g1);
                float r3 = fastrcp(g1 * (g1 + g2));
                o  = -2;                      // band covers jk = NY-3, NY-2, NY-1
                a1 = g1 * r1;                 b1 = -(g1 + g2) * r2; c1 = (2.0f * g1 + g2) * r3;
                a2 = 2.0f * r1;               b2 = -2.0f * r2;      c2 = 2.0f * r3;
            } else {
                float hm = Y[jn_] - Y[jn_ - 1], hp = Y[jn_ + 1] - Y[jn_];
                float s  = hm + hp;
                float q1 = fastrcp(hm * s);
                float q2 = fastrcp(hm * hp);
                float q3 = fastrcp(hp * s);
                o  = -1;                      // band covers jk = jn-1, jn, jn+1
                a1 = -hp * q1;                b1 = (hp - hm) * q2;  c1 = hm * q3;
                a2 = 2.0f * q1;               b2 = -2.0f * q2;      c2 = 2.0f * q3;
            }
        }
        oS[wv][lane] = o;
        ycS[wv][lane][0] = a1; ycS[wv][lane][1] = b1; ycS[wv][lane][2] = c1;
        ycS[wv][lane][3] = a2; ycS[wv][lane][4] = b2; ycS[wv][lane][5] = c2;
    } else {
        // x-direction (periodic) coefficients for row i = i0 + (lane-16)
        int ii = i0 + (lane - 16);
        int im = (ii == 0) ? NXg - 1 : ii - 1;
        int ip = (ii == NXg - 1) ? 0 : ii + 1;
        float hxm = X[ii] - X[im] + ((ii == 0) ? LXg : 0.0f);
        float hxp = X[ip] - X[ii] + ((ii == NXg - 1) ? LXg : 0.0f);
        float sx  = hxm + hxp;
        float q1 = fastrcp(hxm * sx);
        float q2 = fastrcp(hxm * hxp);
        float q3 = fastrcp(hxp * sx);
        xcS[wv][lane - 16][0] = -hxp * q1;
        xcS[wv][lane - 16][1] = (hxp - hxm) * q2;
        xcS[wv][lane - 16][2] = hxm * q3;
        xcS[wv][lane - 16][3] = 2.0f * q1;
        xcS[wv][lane - 16][4] = -2.0f * q2;
        xcS[wv][lane - 16][5] = 2.0f * q3;
    }
    // Same-wave producer/consumer; LDS ops are in-order per wave -> no block barrier.

    // ---------- A matrices: 16x20 f32 field tiles (y-window j0-1 .. j0+18) ----------
    // ISA 16x4 A layout: VGPR c, lanes<16 -> K=4s+c, lanes>=16 -> K=4s+c+2.
    v2f a_u[5], a_v[5], a_p[5], a_pv[5], a_imf[5];
    size_t rowbase = (size_t)(i0 + mn) * NYg;
#pragma unroll
    for (int s = 0; s < 5; ++s) {
#pragma unroll
        for (int c = 0; c < 2; ++c) {
            int kl = 4 * s + c + 2 * hi;
            int jy = iclampi(j0 - 1 + kl, 0, NYg - 1);   // clamped; zero coef outside
            size_t idx = rowbase + (size_t)jy;
            float uu = U[idx], vv = V[idx], pp = P[idx];
            a_u[s][c]   = uu;
            a_v[s][c]   = vv;
            a_p[s][c]   = pp;
            a_pv[s][c]  = pp * vv;
            a_imf[s][c] = fastrcp(MF[idx]);
        }
    }

    // ---------- B matrices: banded FD coefficient matrices C[k, n] from LDS ----------
    int jn = j0 + mn;                 // this lane's output y for columns of B / D
    int ob = oS[wv][mn];
    float ya1 = ycS[wv][mn][0], yb1 = ycS[wv][mn][1], yc1 = ycS[wv][mn][2];
    float ya2 = ycS[wv][mn][3], yb2 = ycS[wv][mn][4], yc2 = ycS[wv][mn][5];
    v2f b1[5], b2[5];
#pragma unroll
    for (int s = 0; s < 5; ++s) {
#pragma unroll
        for (int c = 0; c < 2; ++c) {
            int kl = 4 * s + c + 2 * hi;
            int t  = (kl - 1) - mn - ob;      // jk - jn - o, in-band iff t in {0,1,2}
            b1[s][c] = (t == 0) ? ya1 : (t == 1) ? yb1 : (t == 2) ? yc1 : 0.0f;
            b2[s][c] = (t == 0) ? ya2 : (t == 1) ? yb2 : (t == 2) ? yc2 : 0.0f;
        }
    }

    // ---------- y-direction stencils on the matrix pipe (40x v_wmma_f32_16x16x4_f32) ----------
    v8f d1u = {}, d1v = {}, d1p = {}, d1pv = {}, d1imf = {};
    v8f d2u = {}, d2v = {}, d2p = {};
#pragma unroll
    for (int s = 0; s < 5; ++s) {
        d1u   = WMMA4(a_u[s],   b1[s], d1u);
        d1v   = WMMA4(a_v[s],   b1[s], d1v);
        d1p   = WMMA4(a_p[s],   b1[s], d1p);
        d1pv  = WMMA4(a_pv[s],  b1[s], d1pv);
        d1imf = WMMA4(a_imf[s], b1[s], d1imf);
        d2u   = WMMA4(a_u[s],   b2[s], d2u);
        d2v   = WMMA4(a_v[s],   b2[s], d2v);
        d2p   = WMMA4(a_p[s],   b2[s], d2p);
    }

    // ---------- combine with x-direction stencils (VALU) and store ----------
    // D layout: element r of v8f <-> row M = r + 8*hi, col N = mn.
    int  jj     = jn;
    bool jvalid = (jj < NYg);
    int  jcl    = jvalid ? jj : (NYg - 1);

#pragma unroll
    for (int r = 0; r < 8; ++r) {
        int rr = r + 8 * hi;
        int i  = i0 + rr;
        int im = (i == 0) ? NXg - 1 : i - 1;
        int ip = (i == NXg - 1) ? 0 : i + 1;

        float a1x = xcS[wv][rr][0], b1x = xcS[wv][rr][1], c1x = xcS[wv][rr][2];
        float a2x = xcS[wv][rr][3], b2x = xcS[wv][rr][4], c2x = xcS[wv][rr][5];

        size_t ic  = (size_t)i  * NYg + (size_t)jcl;
        size_t iml = (size_t)im * NYg + (size_t)jcl;
        size_t ipl = (size_t)ip * NYg + (size_t)jcl;

        float uc = U[ic], uxm = U[iml], uxp = U[ipl];
        float vc = V[ic], vxm = V[iml], vxp = V[ipl];
        float pc = P[ic], pxm = P[iml], pxp = P[ipl];
        float mfc = MF[ic], fcc = FC[ic];

        float d1xu  = a1x * uxm + b1x * uc + c1x * uxp;
        float d1xv  = a1x * vxm + b1x * vc + c1x * vxp;
        float d1xp  = a1x * pxm + b1x * pc + c1x * pxp;
        float d1xpu = a1x * (pxm * uxm) + b1x * (pc * uc) + c1x * (pxp * uxp);
        float d2xu  = a2x * uxm + b2x * uc + c2x * uxp;
        float d2xv  = a2x * vxm + b2x * vc + c2x * vxp;
        float d2xp  = a2x * pxm + b2x * pc + c2x * pxp;

        float mf2   = mfc * mfc;
        float m2d1m = mf2 * d1imf[r];
        float fstar = fcc - uc * m2d1m;

        float du = -mfc * uc * d1xu - vc * d1u[r] - mfc * d1xp
                 + NU * (mf2 * d2xu + d2u[r]) + fstar * vc;
        float dv = -mfc * uc * d1xv - vc * d1v[r] - d1p[r]
                 + NU * (mf2 * d2xv + d2v[r]) - fstar * uc;
        float dphi = -mfc * d1xpu - d1pv[r] + NU * (mf2 * d2xp + d2p[r]);

        bool bdry = (jj == 0) || (jj == NYg - 1);
        du = bdry ? 0.0f : du;
        dv = bdry ? 0.0f : dv;

        if (jvalid) {
            size_t o = (size_t)i * NYg + (size_t)jj;
            OUT[o]                   = du;
            OUT[(size_t)NNg + o]     = dv;
            OUT[2 * (size_t)NNg + o] = dphi;
        }
    }
}

extern "C" void kernel_launch(void* const* d_in, const int* in_sizes, int n_in,
                              void* d_out, int out_size, void* d_ws, size_t ws_size,
                              hipStream_t stream) {
    (void)in_sizes; (void)n_in; (void)out_size; (void)d_ws; (void)ws_size;
    const float* U  = (const float*)d_in[0];
    const float* V  = (const float*)d_in[1];
    const float* P  = (const float*)d_in[2];
    const float* X  = (const float*)d_in[3];
    const float* Y  = (const float*)d_in[4];
    const float* MF = (const float*)d_in[5];
    const float* FC = (const float*)d_in[6];
    // d_in[7] = t (unused)
    float* OUT = (float*)d_out;

    const int TJ = (NYg + 15) / 16;   // 46
    const int TI = NXg / 16;          // 90
    const int NT = TI * TJ;           // 4140 waves, 8 waves (256 thr) per block
    dim3 grid((NT + 7) / 8), block(256);
    era5_tend_kernel<<<grid, block, 0, stream>>>(U, V, P, X, Y, MF, FC, OUT);
}